// Img2Prj_34926674051157
// MI455X (gfx1250) — compile-verified
//
#include <hip/hip_runtime.h>
#include <hip/hip_bf16.h>

// Radon forward projection for MI455X (gfx1250, wave32).
//   input : [1, 2, 512, 512] fp32 image
//   output: [1, 2, 768, 736] fp32 sinogram
//
// Strategy (see analysis): gather-bound kernel, image resident in L2.
// Wave = 16 detectors x 4-way t-split; partial sums reduced across lanes
// with a single v_wmma_f32_16x16x4_f32 (ones x partials = column sums).

#define RH 512
#define RW 512
#define RNVIEW 768
#define RNDCT 736
#define RNT 725

typedef __attribute__((ext_vector_type(2))) float v2f;
typedef __attribute__((ext_vector_type(8))) float v8f;

#if defined(__gfx1250__) && __has_builtin(__builtin_amdgcn_wmma_f32_16x16x4_f32)
#define USE_WMMA_RED 1
#else
#define USE_WMMA_RED 0
#endif

// Bilinear sample with zero padding, matching the reference semantics.
__device__ __forceinline__ float radon_sample(const float* __restrict__ img,
                                              float x, float y) {
    float xf = floorf(x);
    float yf = floorf(y);
    int x0 = (int)xf;
    int y0 = (int)yf;
    // Entirely outside the image (all 4 corners invalid): contributes 0.
    if (x0 < -1 || x0 > RW - 1 || y0 < -1 || y0 > RH - 1) return 0.0f;
    float wx = x - xf;
    float wy = y - yf;
    if ((unsigned)x0 < (unsigned)(RW - 1) && (unsigned)y0 < (unsigned)(RH - 1)) {
        // Fast interior path: 4 unmasked loads + lerp.
        const float* p = img + (size_t)y0 * RW + x0;
        float v00 = p[0];
        float v01 = p[1];
        float v10 = p[RW];
        float v11 = p[RW + 1];
        float top = fmaf(wx, v01 - v00, v00);
        float bot = fmaf(wx, v11 - v10, v10);
        return fmaf(wy, bot - top, top);
    }
    // Boundary path: masked, zero-padded gathers.
    int x1 = x0 + 1;
    int y1 = y0 + 1;
    bool vx0 = (unsigned)x0 < (unsigned)RW;
    bool vx1 = (unsigned)x1 < (unsigned)RW;
    bool vy0 = (unsigned)y0 < (unsigned)RH;
    bool vy1 = (unsigned)y1 < (unsigned)RH;
    int xc0 = vx0 ? x0 : 0;
    int xc1 = vx1 ? x1 : 0;
    int yc0 = vy0 ? y0 : 0;
    int yc1 = vy1 ? y1 : 0;
    float v00 = (vx0 && vy0) ? img[(size_t)yc0 * RW + xc0] : 0.0f;
    float v01 = (vx1 && vy0) ? img[(size_t)yc0 * RW + xc1] : 0.0f;
    float v10 = (vx0 && vy1) ? img[(size_t)yc1 * RW + xc0] : 0.0f;
    float v11 = (vx1 && vy1) ? img[(size_t)yc1 * RW + xc1] : 0.0f;
    float top = fmaf(wx, v01 - v00, v00);
    float bot = fmaf(wx, v11 - v10, v10);
    return fmaf(wy, bot - top, top);
}

__global__ __launch_bounds__(256) void radon_fwd_kernel(
    const float* __restrict__ in, float* __restrict__ out) {
    const int lane = threadIdx.x & 31;
    const int waveId = threadIdx.x >> 5;
    const int half = lane >> 4;      // 0: t mod4 in {0,1}; 1: t mod4 in {2,3}
    const int n = lane & 15;         // detector within the wave's 16-tile

    const int view = blockIdx.y;
    const int ch = blockIdx.z;
    const int sIdx = blockIdx.x * 128 + waveId * 16 + n;

    const float* __restrict__ img = in + (size_t)ch * (RH * RW);

    const float th = (float)view * (3.14159265358979323846f / (float)RNVIEW);
    const float cs = cosf(th);
    const float sn = sinf(th);

    const float cx = (RW - 1) * 0.5f;          // 255.5
    const float cy = (RH - 1) * 0.5f;          // 255.5
    const float sVal = (float)sIdx - (float)(RNDCT - 1) * 0.5f;   // s - 367.5
    const float t0 = (float)(half * 2) - (float)(RNT - 1) * 0.5f; // t - 362

    // Ray positions for the two t-slots this lane owns (t0 and t0+1).
    float x0 = fmaf(sVal, cs, fmaf(-t0, sn, cx));
    float y0 = fmaf(sVal, sn, fmaf(t0, cs, cy));
    float x1 = x0 - sn;
    float y1 = y0 + cs;
    const float dX = -4.0f * sn;
    const float dY = 4.0f * cs;

    float acc0 = 0.0f;
    float acc1 = 0.0f;
    for (int ti = half * 2; ti < RNT; ti += 4) {
        acc0 += radon_sample(img, x0, y0);
        if (ti + 1 < RNT) acc1 += radon_sample(img, x1, y1);
        x0 += dX; y0 += dY;
        x1 += dX; y1 += dY;
    }

    // Cross-lane reduction of the 4 per-detector partials.
    // B (4x16 f32, 2 VGPRs, N striped across lane%16) holds the partials;
    // A = ones gives D[m,n] = sum_k B[k,n] independent of the K permutation.
    float sum;
#if USE_WMMA_RED
    {
        v2f a; a[0] = 1.0f; a[1] = 1.0f;
        v2f b; b[0] = acc0; b[1] = acc1;
        v8f c = {};
        v8f d = __builtin_amdgcn_wmma_f32_16x16x4_f32(
            /*neg_a=*/false, a, /*neg_b=*/false, b,
            /*c_mod=*/(short)0, c, /*reuse_a=*/false, /*reuse_b=*/false);
        sum = d[0];   // row M=0 (all rows identical); N = lane%16
    }
#else
    {
        float s01 = acc0 + acc1;
        s01 += __shfl_xor(s01, 16, 32);
        sum = s01;
    }
#endif

    if (half == 0 && sIdx < RNDCT) {
        out[((size_t)ch * RNVIEW + view) * RNDCT + sIdx] = sum;
    }
}

extern "C" void kernel_launch(void* const* d_in, const int* in_sizes, int n_in,
                              void* d_out, int out_size, void* d_ws, size_t ws_size,
                              hipStream_t stream) {
    (void)in_sizes; (void)n_in; (void)d_ws; (void)ws_size; (void)out_size;
    const float* in = (const float*)d_in[0];
    float* out = (float*)d_out;

    dim3 grid((RNDCT + 127) / 128, RNVIEW, 2);  // (6, 768, 2)
    dim3 block(256);                            // 8 waves, wave32
    radon_fwd_kernel<<<grid, block, 0, stream>>>(in, out);
}